// GriddingLayer_40939628266083
// MI455X (gfx1250) — compile-verified
//
#include <hip/hip_runtime.h>

// ---------------------------------------------------------------------------
// GriddingLayer on MI455X (gfx1250):
//   out[b,c,v] = sum_r Wc[c,r] * x[b, r*7+v]
//   Wc[c,r]    = sum_{c': flat(c')==flat(c), region(c')==r} ratio[c']
// GEMM phase (M=5494 cells, N=2048*7=14336 cols, K=17 padded to 20) uses
// V_WMMA_F32_16X16X4_F32: 5 k-steps per 16x16 tile, 8 tiles per wave with
// A-fragments held in registers. Operands are pre-packed slot-major so every
// fragment load is one coalesced global_load_b64 with zero-padding baked in
// (no guards/branches in the inner loop). HBM-bound on the 315MB f32 output.
// ---------------------------------------------------------------------------

typedef __attribute__((ext_vector_type(2))) float v2f;
typedef __attribute__((ext_vector_type(8))) float v8f;

#define GRID_R   82
#define GRID_C   67
#define N_VARS   7
#define N_REG    17
#define N_CELLS  5494
#define BATCH    2048

#define KPAD     20                 // K=17 padded to 5 k-steps of 4
#define K_SLOTS  10                 // KPAD/2 slot pairs {2s, 2s+1}
#define MPAD     5504               // 344*16
#define M_TILES  344
#define N_COLS   (BATCH * N_VARS)   // 14336 = 896*16 exactly
#define N_TILES  896
#define WAVES_PER_BLOCK 4
#define NT_PER_WAVE     8           // N-tiles per wave (A reuse in registers)

// ---------------- helper kernels ----------------

__global__ void zero_f32_kernel(float* __restrict__ p, int n) {
    int i = blockIdx.x * blockDim.x + threadIdx.x;
    if (i < n) p[i] = 0.0f;
}

__global__ void scatter_w_kernel(const float* __restrict__ ratio,
                                 const int* __restrict__ region_id,
                                 const int* __restrict__ rows,
                                 const int* __restrict__ cols,
                                 float* __restrict__ W) {
    int c = blockIdx.x * blockDim.x + threadIdx.x;
    if (c < N_CELLS) {
        int g = rows[c] * GRID_C + cols[c];
        atomicAdd(&W[g * N_REG + region_id[c]], ratio[c]);
    }
}

// Wcs[s][m] = { W[flat(m), 2s], W[flat(m), 2s+1] }, zero-padded (m>=N_CELLS or r>=17).
__global__ void gather_wcs_kernel(const int* __restrict__ rows,
                                  const int* __restrict__ cols,
                                  const float* __restrict__ W,
                                  float* __restrict__ Wcs) {
    int t = blockIdx.x * blockDim.x + threadIdx.x;   // over K_SLOTS*MPAD
    if (t >= K_SLOTS * MPAD) return;
    int s = t / MPAD;
    int m = t - s * MPAD;
    float v0 = 0.0f, v1 = 0.0f;
    if (m < N_CELLS) {
        int g = rows[m] * GRID_C + cols[m];
        int r0 = 2 * s;
        if (r0     < N_REG) v0 = W[g * N_REG + r0];
        if (r0 + 1 < N_REG) v1 = W[g * N_REG + r0 + 1];
    }
    Wcs[2 * t]     = v0;
    Wcs[2 * t + 1] = v1;
}

// Xpp[s][n] = { x[b, 2s, v], x[b, 2s+1, v] } with n = b*7+v, zero-padded r>=17.
__global__ void pack_x_kernel(const float* __restrict__ x,
                              float* __restrict__ Xpp) {
    int t = blockIdx.x * blockDim.x + threadIdx.x;   // over K_SLOTS*N_COLS
    if (t >= K_SLOTS * N_COLS) return;
    int s = t / N_COLS;
    int n = t - s * N_COLS;
    int b = n / N_VARS;
    int v = n - b * N_VARS;
    int r0 = 2 * s;
    const float* xb = x + (size_t)b * (N_REG * N_VARS) + v;
    float v0 = (r0     < N_REG) ? xb[r0 * N_VARS]       : 0.0f;
    float v1 = (r0 + 1 < N_REG) ? xb[(r0 + 1) * N_VARS] : 0.0f;
    Xpp[2 * t]     = v0;
    Xpp[2 * t + 1] = v1;
}

// ---------------- WMMA GEMM ----------------
// Fragment layouts for V_WMMA_F32_16X16X4_F32 (wave32), per ISA 7.12.2:
//   A 16x4: lane l<16 -> A[l, {k0,k0+1}]; lane l>=16 -> A[l-16, {k0+2,k0+3}]
//   B 4x16: lane l<16 -> B[{k0,k0+1}, l]; lane l>=16 -> B[{k0+2,k0+3}, l-16]
//   C/D 16x16: VGPR i -> M=i (lanes 0-15), M=i+8 (lanes 16-31), N=lane&15
// Slot index s = 2*kstep + (lane>=16) selects the packed K-pair {2s,2s+1}.

__global__ __launch_bounds__(WAVES_PER_BLOCK * 32)
void wmma_gemm_kernel(const float* __restrict__ Xpp,   // (K_SLOTS, N_COLS, 2)
                      const float* __restrict__ Wcs,   // (K_SLOTS, MPAD, 2)
                      float* __restrict__ out) {       // (BATCH, N_CELLS, N_VARS)
    const int lane   = threadIdx.x & 31;
    const int wave   = threadIdx.x >> 5;
    const int half   = lane >> 4;
    const int l15    = lane & 15;
    const int mtile  = blockIdx.y;
    const int c0     = mtile * 16;
    const int ntile0 = (blockIdx.x * WAVES_PER_BLOCK + wave) * NT_PER_WAVE;

    const v2f* Af = (const v2f*)Wcs;
    const v2f* Bf = (const v2f*)Xpp;

    // A fragments: one coalesced b64 per k-step, reused for all 8 N-tiles.
    v2f afrag[5];
#pragma unroll
    for (int kk = 0; kk < 5; ++kk)
        afrag[kk] = Af[(size_t)(2 * kk + half) * MPAD + (c0 + l15)];

    const bool full_tile = (c0 + 16 <= N_CELLS);   // uniform: scalar branch

#pragma unroll
    for (int t = 0; t < NT_PER_WAVE; ++t) {
        const int n = (ntile0 + t) * 16 + l15;     // this lane's output column
        const int b = n / N_VARS;
        const int v = n - b * N_VARS;

        v8f acc = {0.f, 0.f, 0.f, 0.f, 0.f, 0.f, 0.f, 0.f};
#pragma unroll
        for (int kk = 0; kk < 5; ++kk) {
            v2f bfrag = Bf[(size_t)(2 * kk + half) * N_COLS + n];
            acc = __builtin_amdgcn_wmma_f32_16x16x4_f32(
                /*neg_a=*/false, afrag[kk], /*neg_b=*/false, bfrag,
                /*c_mod=*/(short)0, acc, /*reuse_a=*/false, /*reuse_b=*/false);
        }

        float* obase = out + (size_t)b * (N_CELLS * N_VARS) + v
                           + (size_t)(c0 + (half << 3)) * N_VARS;
        if (full_tile) {
#pragma unroll
            for (int i = 0; i < 8; ++i) obase[i * N_VARS] = acc[i];
        } else {
#pragma unroll
            for (int i = 0; i < 8; ++i)
                if (c0 + (half << 3) + i < N_CELLS) obase[i * N_VARS] = acc[i];
        }
    }
}

// ---------------- launch ----------------

extern "C" void kernel_launch(void* const* d_in, const int* in_sizes, int n_in,
                              void* d_out, int out_size, void* d_ws, size_t ws_size,
                              hipStream_t stream) {
    const float* x         = (const float*)d_in[0];  // (2048, 119)
    const float* ratio     = (const float*)d_in[1];  // (5494,)
    const int*   region_id = (const int*)  d_in[2];  // (5494,)
    const int*   rows      = (const int*)  d_in[3];  // (5494,)
    const int*   cols      = (const int*)  d_in[4];  // (5494,)
    float*       out       = (float*)d_out;          // (2048, 5494, 1, 7)

    char* ws = (char*)d_ws;
    float* W   = (float*)(ws);                       // 5494*17 f32   (~374 KB)
    float* Wcs = (float*)(ws + (512u << 10));        // 10*5504*2 f32 (~440 KB)
    float* Xpp = (float*)(ws + (1024u << 10));       // 10*14336*2 f32 (~1.1 MB)

    const int wcount = N_CELLS * N_REG;              // 93398
    zero_f32_kernel<<<(wcount + 255) / 256, 256, 0, stream>>>(W, wcount);

    scatter_w_kernel<<<(N_CELLS + 255) / 256, 256, 0, stream>>>(
        ratio, region_id, rows, cols, W);

    const int gcount = K_SLOTS * MPAD;               // 55040
    gather_wcs_kernel<<<(gcount + 255) / 256, 256, 0, stream>>>(rows, cols, W, Wcs);

    const int pcount = K_SLOTS * N_COLS;             // 143360
    pack_x_kernel<<<(pcount + 255) / 256, 256, 0, stream>>>(x, Xpp);

    dim3 grid(N_TILES / (WAVES_PER_BLOCK * NT_PER_WAVE), M_TILES);  // (28, 344)
    dim3 block(WAVES_PER_BLOCK * 32);                               // 4 waves
    wmma_gemm_kernel<<<grid, block, 0, stream>>>(Xpp, Wcs, out);
}